// disan_32263794327663
// MI455X (gfx1250) — compile-verified
//
#include <hip/hip_runtime.h>

typedef float v2f __attribute__((ext_vector_type(2)));
typedef float v8f __attribute__((ext_vector_type(8)));

constexpr int BS  = 8;
constexpr int SL  = 256;
constexpr int VEC = 256;                 // K == N == VEC for every GEMM here
constexpr int MROWS = BS * SL;           // 2048
constexpr int GI  = 4;                   // i-rows produced per attention block

// ---------------------------------------------------------------------------
// K=256 WMMA accumulation for one 16x16 f32 tile (V_WMMA_F32_16X16X4_F32).
// Compile-time strides -> pointer-march + immediate-offset loads (no u64 muls).
// A frag: lanes 0-15 rows M, K=k0+{0,1}; lanes 16-31 K=k0+{2,3}. B mirrored.
// ---------------------------------------------------------------------------
__device__ inline v8f wmma_accum256(const float* __restrict__ A,
                                    const float* __restrict__ B,
                                    int tm, int tn, int l15, int grp, v8f acc) {
  const float* ap = A + (size_t)(tm + l15) * VEC + 2 * grp;
  const float* bp = B + (size_t)(2 * grp) * VEC + (tn + l15);
#pragma unroll 4
  for (int k0 = 0; k0 < VEC; k0 += 4) {
    v2f a = *reinterpret_cast<const v2f*>(ap);   // 8B-aligned b64 load
    v2f b;
    b.x = bp[0];
    b.y = bp[VEC];
    ap += 4;
    bp += 4 * VEC;
    acc = __builtin_amdgcn_wmma_f32_16x16x4_f32(false, a, false, b,
                                                (short)0, acc, false, false);
  }
  return acc;
}

// ---------------------------------------------------------------------------
// GEMM + fused epilogue. One wave owns one 16x16 C tile (EXEC uniform all-1s).
//   MODE 0: ELU(acc + b0)                               -> rep_map
//   MODE 1: acc + b0(optional)                          -> dependent / head+b_f
//   MODE 2: g = sigmoid(acc + b0+b1+b2); g*rep+(1-g)*at -> fusion gate output
//   DUAL : acc = A*B + A2*B2 (K-loop run twice)
// ---------------------------------------------------------------------------
template <int MODE, bool DUAL>
__global__ __launch_bounds__(256) void gemm_kernel(
    const float* __restrict__ A,  const float* __restrict__ B,
    const float* __restrict__ A2, const float* __restrict__ B2,
    const float* __restrict__ bias0, const float* __restrict__ bias1,
    const float* __restrict__ bias2,
    const float* __restrict__ repP, const float* __restrict__ attnP,
    float* __restrict__ C) {
  const int lane = threadIdx.x & 31;
  const int wave = threadIdx.x >> 5;
  constexpr int NTN = VEC / 16;                  // 16 N-tiles
  const int tile = blockIdx.x * 8 + wave;        // exact grid, no bounds check
  const int tm   = (tile / NTN) << 4;
  const int tn   = (tile % NTN) << 4;
  const int grp  = lane >> 4;
  const int l15  = lane & 15;

  v8f acc = {0.f, 0.f, 0.f, 0.f, 0.f, 0.f, 0.f, 0.f};
  acc = wmma_accum256(A, B, tm, tn, l15, grp, acc);
  if constexpr (DUAL) acc = wmma_accum256(A2, B2, tm, tn, l15, grp, acc);

  const int ncol = tn + l15;
  float bs = 0.f;
  if (bias0) bs += bias0[ncol];
  if (bias1) bs += bias1[ncol];
  if (bias2) bs += bias2[ncol];

  float* crow = C + (size_t)(tm + grp * 8) * VEC + ncol;   // D: VGPR v -> M=v(+8)
#pragma unroll
  for (int v = 0; v < 8; ++v) {
    float val = acc[v] + bs;
    if constexpr (MODE == 0) {
      val = (val > 0.f) ? val : (__expf(val) - 1.f);        // ELU
    } else if constexpr (MODE == 2) {
      const size_t idx = (size_t)(tm + grp * 8 + v) * VEC + ncol;
      float g = __builtin_amdgcn_rcpf(1.f + __expf(-val));  // sigmoid
      val = g * repP[idx] + (1.f - g) * attnP[idx];
    }
    crow[(size_t)v * VEC] = val;                            // imm-offset stores
  }
}

// ---------------------------------------------------------------------------
// Fused masked softmax-attention. logit = 5*tanh((dep[j,d]+head[i,d]+b_f)/5)
// is bounded in [-5,5] -> softmax needs no running max: single pass
// p = exp(l)*mask, s += p, acc += p*rep.  tanh via one exp + one rcp.
// Each block produces GI consecutive i-rows so every dep/rep row load is
// amortized over GI updates (L2 traffic /GI) -> firmly trans-op bound.
// ---------------------------------------------------------------------------
__global__ __launch_bounds__(256) void attn_kernel(
    const float* __restrict__ dep, const float* __restrict__ headb,
    const float* __restrict__ rep, const unsigned char* __restrict__ mask,
    float* __restrict__ attn) {
  const int d  = threadIdx.x;
  constexpr int IBLK = SL / GI;                  // 64 i-blocks per batch
  const int b  = blockIdx.x / IBLK;
  const int i0 = (blockIdx.x % IBLK) * GI;

  const float* depp = dep + (size_t)b * SL * VEC + d;
  const float* repp = rep + (size_t)b * SL * VEC + d;
  const unsigned char* mb = mask + (size_t)b * SL;

  float hb[GI], s[GI], acc[GI];
#pragma unroll
  for (int g = 0; g < GI; ++g) {
    hb[g]  = headb[((size_t)b * SL + i0 + g) * VEC + d];   // includes +b_f
    s[g]   = 0.f;
    acc[g] = 0.f;
  }

  // main phase: rows j < i0 feed all GI outputs (branch-free, mask as mult)
#pragma unroll 2
  for (int j = 0; j < i0; ++j) {
    const float m  = (float)mb[j];
    const float dv = depp[(size_t)j * VEC];
    const float rv = repp[(size_t)j * VEC];
#pragma unroll
    for (int g = 0; g < GI; ++g) {
      const float u = __expf(0.4f * (dv + hb[g]));                // e^{2x/5}
      const float l = 5.f - 10.f * __builtin_amdgcn_rcpf(u + 1.f);// 5*tanh(x/5)
      const float p = __expf(l) * m;               // bounded, no overflow
      s[g]   += p;
      acc[g] += p * rv;
    }
  }

  // causal tail: j in [i0, i0+GI-2]; output g active iff j < i0+g
#pragma unroll
  for (int t = 0; t < GI - 1; ++t) {
    const int j = i0 + t;
    const float m  = (float)mb[j];
    const float dv = depp[(size_t)j * VEC];
    const float rv = repp[(size_t)j * VEC];
#pragma unroll
    for (int g = t + 1; g < GI; ++g) {
      const float u = __expf(0.4f * (dv + hb[g]));
      const float l = 5.f - 10.f * __builtin_amdgcn_rcpf(u + 1.f);
      const float p = __expf(l) * m;
      s[g]   += p;
      acc[g] += p * rv;
    }
  }

#pragma unroll
  for (int g = 0; g < GI; ++g) {
    const size_t idx = ((size_t)b * SL + i0 + g) * VEC + d;
    attn[idx] = (s[g] > 0.f) ? acc[g] * __builtin_amdgcn_rcpf(s[g]) : 0.f;
  }
}

// ---------------------------------------------------------------------------
// inputs order: inputs, rep_mask, W_h, b_h, W_f1, W_f2, b_f,
//               W_fg1, W_fg2, b_fg1, b_fg2, b_fg3
// Workspace: rep | dep | head+b_f | attn  (4 x 2MB f32, L2-resident)
// ---------------------------------------------------------------------------
extern "C" void kernel_launch(void* const* d_in, const int* in_sizes, int n_in,
                              void* d_out, int out_size, void* d_ws, size_t ws_size,
                              hipStream_t stream) {
  (void)in_sizes; (void)n_in; (void)out_size; (void)ws_size;

  const float* inputs        = (const float*)d_in[0];
  const unsigned char* rmask = (const unsigned char*)d_in[1];
  const float* W_h   = (const float*)d_in[2];
  const float* b_h   = (const float*)d_in[3];
  const float* W_f1  = (const float*)d_in[4];
  const float* W_f2  = (const float*)d_in[5];
  const float* b_f   = (const float*)d_in[6];
  const float* W_fg1 = (const float*)d_in[7];
  const float* W_fg2 = (const float*)d_in[8];
  const float* b_fg1 = (const float*)d_in[9];
  const float* b_fg2 = (const float*)d_in[10];
  const float* b_fg3 = (const float*)d_in[11];
  float* out = (float*)d_out;

  const size_t plane = (size_t)MROWS * VEC;
  float* ws    = (float*)d_ws;
  float* rep   = ws;
  float* dep   = ws + plane;
  float* headb = ws + 2 * plane;
  float* attn  = ws + 3 * plane;

  constexpr int GEMM_BLOCKS = (MROWS / 16) * (VEC / 16) / 8;  // 256 (8 waves/blk)
  dim3 gblock(256);

  // rep = ELU(inputs*W_h + b_h)
  gemm_kernel<0, false><<<dim3(GEMM_BLOCKS), gblock, 0, stream>>>(
      inputs, W_h, nullptr, nullptr, b_h, nullptr, nullptr, nullptr, nullptr, rep);
  // dep = rep*W_f1
  gemm_kernel<1, false><<<dim3(GEMM_BLOCKS), gblock, 0, stream>>>(
      rep, W_f1, nullptr, nullptr, nullptr, nullptr, nullptr, nullptr, nullptr, dep);
  // headb = rep*W_f2 + b_f
  gemm_kernel<1, false><<<dim3(GEMM_BLOCKS), gblock, 0, stream>>>(
      rep, W_f2, nullptr, nullptr, b_f, nullptr, nullptr, nullptr, nullptr, headb);
  // fused masked softmax attention (GI rows per block)
  attn_kernel<<<dim3(BS * (SL / GI)), gblock, 0, stream>>>(
      dep, headb, rep, rmask, attn);
  // out = sigmoid(rep*W_fg1 + attn*W_fg2 + b1+b2+b3)*rep + (1-g)*attn
  gemm_kernel<2, true><<<dim3(GEMM_BLOCKS), gblock, 0, stream>>>(
      rep, W_fg1, attn, W_fg2, b_fg1, b_fg2, b_fg3, rep, attn, out);
}